// hash_embedding_33809982554502
// MI455X (gfx1250) — compile-verified
//
#include <hip/hip_runtime.h>
#include <math.h>
#include <stdint.h>

// ---------------------------------------------------------------------------
// Instant-NGP multires hash encoding, CDNA5 (gfx1250).
//
// Roofline: 64 MB of embedding tables fit in the 192 MB L2 -> the 134M random
// float2 gathers are L2-resident; output (128 MB, streamed once) is written
// with non-temporal stores so it cannot evict the tables. x coords are staged
// per-block into LDS with gfx1250 async global->LDS loads (ASYNCcnt path),
// since 16 threads (one per level) share each point's coordinates.
// ---------------------------------------------------------------------------

#define TBL_SIZE   (1u << 19)          // hash table entries per level
#define TBL_MASK   (TBL_SIZE - 1u)
#define NLVL       16
#define PTS_PER_BLK 16
#define THREADS    256                  // 8 wave32 per block

typedef float v2f __attribute__((ext_vector_type(2)));

struct LevelRes { float r[NLVL]; };

__global__ __launch_bounds__(THREADS)
void hashgrid_encode(const float* __restrict__ x,
                     const float* __restrict__ bbox,
                     const float* __restrict__ emb,
                     float* __restrict__ out,
                     LevelRes res, int n_points)
{
    __shared__ float sx[PTS_PER_BLK * 3];   // raw coords of this block's 16 points

    const int tid = threadIdx.x;
    const int blk = blockIdx.x;

    // ---- async stage: 48 dwords of x -> LDS (gfx1250 ASYNC path) ----------
    if (tid < PTS_PER_BLK * 3 && (blk * (PTS_PER_BLK * 3) + tid) < n_points * 3) {
        uint32_t lds_addr = (uint32_t)(uintptr_t)(&sx[tid]);              // LDS byte offset
        uint32_t goff     = (uint32_t)((blk * (PTS_PER_BLK * 3) + tid) * 4);
        asm volatile("global_load_async_to_lds_b32 %0, %1, %2"
                     :: "v"(lds_addr), "v"(goff), "s"(x)
                     : "memory");
    }

    // bounding box is uniform across the grid -> scalar loads, overlapped
    // with the async copy above.
    const float mnx = bbox[0], mny = bbox[1], mnz = bbox[2];
    const float dx  = bbox[3] - mnx;
    const float dy  = bbox[4] - mny;
    const float dz  = bbox[5] - mnz;

    asm volatile("s_wait_asynccnt 0x0" ::: "memory");
    __syncthreads();

    const int lp = tid >> 4;        // local point 0..15
    const int l  = tid & 15;        // resolution level 0..15
    const int n  = blk * PTS_PER_BLK + lp;
    if (n >= n_points) return;

    // normalized coords, scaled by this level's resolution
    const float Nl = res.r[l];
    const float px = ((sx[lp * 3 + 0] - mnx) / dx) * Nl;
    const float py = ((sx[lp * 3 + 1] - mny) / dy) * Nl;
    const float pz = ((sx[lp * 3 + 2] - mnz) / dz) * Nl;

    const float cx = floorf(px), cy = floorf(py), cz = floorf(pz);
    const float fx = px - cx,   fy = py - cy,   fz = pz - cz;
    const int   ix = (int)cx,   iy = (int)cy,   iz = (int)cz;

    // NGP spatial hash: (x*1) ^ (y*2654435761) ^ (z*805459861), uint32 wrap
    const uint32_t hx0 = (uint32_t)ix;
    const uint32_t hx1 = (uint32_t)(ix + 1);
    const uint32_t hy0 = (uint32_t)iy       * 2654435761u;
    const uint32_t hy1 = (uint32_t)(iy + 1) * 2654435761u;
    const uint32_t hz0 = (uint32_t)iz       * 805459861u;
    const uint32_t hz1 = (uint32_t)(iz + 1) * 805459861u;

    const uint32_t hx[2] = { hx0, hx1 };
    const uint32_t hy[2] = { hy0, hy1 };
    const uint32_t hz[2] = { hz0, hz1 };

    // per-level table slice; 32-bit byte offsets keep the saddr-form loads
    const char* tbl = (const char*)emb + (size_t)l * (TBL_SIZE * 8u);

    // corner order matches OFFSETS: i (x) slowest, k (z) fastest
    uint32_t off[8];
#pragma unroll
    for (int c = 0; c < 8; ++c) {
        const int i = (c >> 2) & 1, j = (c >> 1) & 1, k = c & 1;
        off[c] = ((hx[i] ^ hy[j] ^ hz[k]) & TBL_MASK) * 8u;
    }

    // issue all 8 gathers back-to-back (8 global_load_b64 in flight)
    v2f f[8];
#pragma unroll
    for (int c = 0; c < 8; ++c)
        f[c] = *(const v2f*)(tbl + off[c]);

    const float wx[2] = { 1.0f - fx, fx };
    const float wy[2] = { 1.0f - fy, fy };
    const float wz[2] = { 1.0f - fz, fz };

    float ax = 0.0f, ay = 0.0f;
#pragma unroll
    for (int c = 0; c < 8; ++c) {
        const int i = (c >> 2) & 1, j = (c >> 1) & 1, k = c & 1;
        const float w = wx[i] * wy[j] * wz[k];
        ax = fmaf(w, f[c].x, ax);
        ay = fmaf(w, f[c].y, ay);
    }

    // out[n, l*2 : l*2+2]; consecutive lanes -> consecutive float2 -> fully
    // coalesced. Non-temporal so the 128 MB output stream does not evict the
    // L2-resident embedding tables.
    v2f o = { ax, ay };
    __builtin_nontemporal_store(o, (v2f*)out + ((size_t)n * NLVL + l));
}

extern "C" void kernel_launch(void* const* d_in, const int* in_sizes, int n_in,
                              void* d_out, int out_size, void* d_ws, size_t ws_size,
                              hipStream_t stream) {
    const float* x    = (const float*)d_in[0];   // [N,3]
    const float* bbox = (const float*)d_in[1];   // [2,3]
    const float* emb  = (const float*)d_in[2];   // [L,T,F]
    float*       out  = (float*)d_out;           // [N,L*F]

    const int n_points = in_sizes[0] / 3;

    // Level resolutions: floor(16 * b^l), b = exp((ln512 - ln16)/15) in f32.
    // Replicate XLA's integer_pow (square-and-multiply) rounding so levels
    // that land exactly on integers (l = 3,6,9,12,15) floor identically.
    LevelRes res;
    const float b = expf((logf(512.0f) - logf(16.0f)) / 15.0f);
    for (int l = 0; l < NLVL; ++l) {
        float acc = 1.0f, base = b;
        int e = l;
        while (e) {
            if (e & 1) acc *= base;
            e >>= 1;
            if (e) base *= base;
        }
        res.r[l] = floorf(16.0f * acc);
    }

    const int blocks = (n_points + PTS_PER_BLK - 1) / PTS_PER_BLK;
    hashgrid_encode<<<blocks, THREADS, 0, stream>>>(x, bbox, emb, out, res, n_points);
}